// LSTM_61753039782564
// MI455X (gfx1250) — compile-verified
//
#include <hip/hip_runtime.h>
#include <hip/hip_bf16.h>

#define T_LEN 262144
#define IN_SZ 15
#define HID   16
#define P_OUT 10
#define CH    128                      // timesteps per async chunk
#define NC    (T_LEN / CH)             // 2048 chunks
#define XPAD  16
#define XBUF_FLOATS (CH * IN_SZ + XPAD)
#define CHUNK_BYTES (CH * IN_SZ * 4)   // 7680

typedef __attribute__((ext_vector_type(2))) float v2f;
typedef __attribute__((ext_vector_type(8))) float v8f;

__device__ __forceinline__ float fast_sigmoid(float x) {
    float e = __builtin_amdgcn_exp2f(-1.4426950408889634f * x);
    return __builtin_amdgcn_rcpf(1.0f + e);
}
__device__ __forceinline__ float fast_tanh(float x) {
    float e = __builtin_amdgcn_exp2f(2.8853900817779268f * x);  // 2^(2x*log2e)
    return 1.0f - 2.0f * __builtin_amdgcn_rcpf(e + 1.0f);
}

// Generic pointers into LDS carry the LDS byte offset in their low 32 bits
// (aperture lives in the high 32 bits on AMDGPU).
__device__ __forceinline__ unsigned lds_off(const float* p) {
    return (unsigned)(unsigned long long)(size_t)p;
}

// Stream one chunk of x (CH*IN_SZ floats, contiguous) into LDS with the
// CDNA5 async copy engine: 15 x b128, 32 lanes x 16B each = 512B/instr.
__device__ __forceinline__ void issue_chunk_async(const float* x, int c,
                                                  const float* ldsbuf, int lane) {
    unsigned l = lds_off(ldsbuf) + (unsigned)lane * 16u;
    const char* g = (const char*)x + (size_t)c * CHUNK_BYTES + lane * 16;
#pragma unroll
    for (int j = 0; j < CHUNK_BYTES / 512; ++j) {
        asm volatile("global_load_async_to_lds_b128 %0, %1, off offset:%2"
                     :: "v"(l), "v"(g), "i"(j * 512) : "memory");
    }
}

__global__ __launch_bounds__(32, 1) void lstm2_seq_kernel(
    const float* __restrict__ x,
    const float* __restrict__ W_ih0, const float* __restrict__ W_hh0,
    const float* __restrict__ b_ih0, const float* __restrict__ b_hh0,
    const float* __restrict__ W_ih1, const float* __restrict__ W_hh1,
    const float* __restrict__ b_ih1, const float* __restrict__ b_hh1,
    const float* __restrict__ W_reg, const float* __restrict__ b_reg,
    float* __restrict__ out)
{
    __shared__ __align__(16) float xbuf[2][XBUF_FLOATS];
    __shared__ __align__(16) float hbuf[32];   // [0..15]=h0, [16..31]=h1

    const int  lane = (int)threadIdx.x;
    const bool L0   = lane < 16;
    const int  unit = lane & 15;

    // ---- load per-lane weight rows into VGPRs (uniform instruction stream) ----
    const float* WaSrc = L0 ? W_hh0 : W_ih1;   // 64x16 row-major
    const float* WbSrc = L0 ? W_ih0 : W_hh1;   // L0: 64x15, L1: 64x16
    const float* BiA   = L0 ? b_ih0 : b_ih1;
    const float* BiB   = L0 ? b_hh0 : b_hh1;
    const int    strB  = L0 ? IN_SZ : HID;

    v2f wa[4][8], wb[4][8];
    float bias[4];
#pragma unroll
    for (int gq = 0; gq < 4; ++gq) {           // PyTorch gate order i,f,g,o
        const int row = gq * 16 + unit;
#pragma unroll
        for (int k = 0; k < 16; ++k) {
            float av = WaSrc[row * HID + k];
            const bool zb = L0 && (k == IN_SZ);      // pad col 15 of W_ih0 with 0
            const int  kb = zb ? 0 : k;              // keep load in-bounds
            float bv = WbSrc[row * strB + kb];
            bv = zb ? 0.0f : bv;
            if (k & 1) { wa[gq][k >> 1].y = av; wb[gq][k >> 1].y = bv; }
            else       { wa[gq][k >> 1].x = av; wb[gq][k >> 1].x = bv; }
        }
        bias[gq] = BiA[row] + BiB[row];
    }

    // ---- init state ----
    hbuf[lane] = 0.0f;
    if (lane < XPAD) { xbuf[0][CH * IN_SZ + lane] = 0.0f; xbuf[1][CH * IN_SZ + lane] = 0.0f; }
    float cst = 0.0f, hst = 0.0f;
    const float* ha  = &hbuf[0];
    const float* h1b = &hbuf[16];

    issue_chunk_async(x, 0, &xbuf[0][0], lane);

    // One pass = layer0 at time p (lanes 0-15) + layer1 at time p-1 (lanes 16-31).
    auto lstm_pass = [&](const float* hbp, bool cmt) {
        v2f A0 = {bias[0], 0.f}, A1 = {bias[1], 0.f}, A2 = {bias[2], 0.f}, A3 = {bias[3], 0.f};
        v2f B0 = {0.f, 0.f},     B1 = {0.f, 0.f},     B2 = {0.f, 0.f},     B3 = {0.f, 0.f};
#pragma unroll
        for (int kk = 0; kk < 8; ++kk) {
            v2f hav = { ha[2 * kk],  ha[2 * kk + 1]  };   // aligned LDS broadcast
            v2f hbv = { hbp[2 * kk], hbp[2 * kk + 1] };   // x row (stride 15) or h1
            A0 = __builtin_elementwise_fma(wa[0][kk], hav, A0);
            A1 = __builtin_elementwise_fma(wa[1][kk], hav, A1);
            A2 = __builtin_elementwise_fma(wa[2][kk], hav, A2);
            A3 = __builtin_elementwise_fma(wa[3][kk], hav, A3);
            B0 = __builtin_elementwise_fma(wb[0][kk], hbv, B0);
            B1 = __builtin_elementwise_fma(wb[1][kk], hbv, B1);
            B2 = __builtin_elementwise_fma(wb[2][kk], hbv, B2);
            B3 = __builtin_elementwise_fma(wb[3][kk], hbv, B3);
        }
        float gi = (A0.x + A0.y) + (B0.x + B0.y);
        float gf = (A1.x + A1.y) + (B1.x + B1.y);
        float gg = (A2.x + A2.y) + (B2.x + B2.y);
        float go = (A3.x + A3.y) + (B3.x + B3.y);
        float si = fast_sigmoid(gi), sf = fast_sigmoid(gf);
        float tg = fast_tanh(gg),    so = fast_sigmoid(go);
        float cn = sf * cst + si * tg;
        float hn = so * fast_tanh(cn);
        cst = cmt ? cn : cst;
        hst = cmt ? hn : hst;
        hbuf[lane] = hst;                 // publish (same-wave DS ops are in-order)
    };

    int p = 0;
#pragma unroll 1
    for (int c = 0; c < NC; ++c) {
        if (c + 1 < NC) {
            issue_chunk_async(x, c + 1, &xbuf[(c + 1) & 1][0], lane);
            asm volatile("s_wait_asynccnt 0xf" ::: "memory");  // chunk c complete
        } else {
            asm volatile("s_wait_asynccnt 0x0" ::: "memory");
        }
        const float* xb = &xbuf[c & 1][0];
#pragma unroll 1
        for (int s = 0; s < CH; ++s, ++p) {
            const float* hbp = L0 ? (xb + s * IN_SZ) : h1b;
            bool cmt = L0 || (p > 0);     // layer1 half skips the priming pass
            lstm_pass(hbp, cmt);
        }
    }
    // Drain pass p == T: layer1 finishes t = T-1; layer0 half does not commit.
    lstm_pass(L0 ? ha : h1b, !L0);

    // ---- head: out = W_reg @ h1 + b_reg via v_wmma_f32_16x16x4_f32 (K-chained) ----
    // A 16x4 f32 layout: lanes 0-15 hold K=0(.x),K=1(.y); lanes 16-31 hold K=2,K=3.
    // B 4x16: row K striped across lanes within a VGPR (h broadcast to all N).
    // C/D:    lane n<16 -> D[v][n], lane 16+n -> D[v+8][n].
    const int koff = (lane >> 4) * 2;
    v8f acc = {0.f, 0.f, 0.f, 0.f, 0.f, 0.f, 0.f, 0.f};
#pragma unroll
    for (int kc = 0; kc < 4; ++kc) {
        const int kb = kc * 4 + koff;
        float w0 = 0.f, w1 = 0.f;
        if (unit < P_OUT) { w0 = W_reg[unit * HID + kb]; w1 = W_reg[unit * HID + kb + 1]; }
        v2f a = { w0, w1 };
        v2f b = { hbuf[16 + kb], hbuf[16 + kb + 1] };
        acc = __builtin_amdgcn_wmma_f32_16x16x4_f32(false, a, false, b,
                                                    (short)0, acc, false, false);
    }
    if (lane == 0) {
#pragma unroll
        for (int pp = 0; pp < 8; ++pp) out[pp] = acc[pp] + b_reg[pp];
    }
    if (lane == 16) {
        out[8] = acc[0] + b_reg[8];
        out[9] = acc[1] + b_reg[9];
    }
}

extern "C" void kernel_launch(void* const* d_in, const int* in_sizes, int n_in,
                              void* d_out, int out_size, void* d_ws, size_t ws_size,
                              hipStream_t stream) {
    (void)in_sizes; (void)n_in; (void)d_ws; (void)ws_size; (void)out_size;
    const float* x     = (const float*)d_in[0];
    const float* W_ih0 = (const float*)d_in[1];
    const float* W_hh0 = (const float*)d_in[2];
    const float* b_ih0 = (const float*)d_in[3];
    const float* b_hh0 = (const float*)d_in[4];
    const float* W_ih1 = (const float*)d_in[5];
    const float* W_hh1 = (const float*)d_in[6];
    const float* b_ih1 = (const float*)d_in[7];
    const float* b_hh1 = (const float*)d_in[8];
    const float* W_reg = (const float*)d_in[9];
    const float* b_reg = (const float*)d_in[10];
    float* out = (float*)d_out;

    lstm2_seq_kernel<<<dim3(1), dim3(32), 0, stream>>>(
        x, W_ih0, W_hh0, b_ih0, b_hh0, W_ih1, W_hh1, b_ih1, b_hh1, W_reg, b_reg, out);
}